// HGTBaseline_24816321036925
// MI455X (gfx1250) — compile-verified
//
#include <hip/hip_runtime.h>
#include <math.h>
#include <float.h>

// ---------------------------------------------------------------------------
// HGT forward for gfx1250 (MI455X). fp32 end-to-end; dense GEMMs use
// V_WMMA_F32_16X16X4_F32 with the B operand staged through LDS in
// pair-interleaved form so each B fragment is a single ds_load_b64.
// Each wave computes a 16x64 strip with 4 accumulators (every A fragment
// feeds 4 WMMAs); A loads are 2-stage software pipelined.
// ---------------------------------------------------------------------------

typedef float v2f __attribute__((ext_vector_type(2)));
typedef float v8f __attribute__((ext_vector_type(8)));

__device__ __forceinline__ float gelu_exact(float x) {
    return 0.5f * x * (1.0f + erff(x * 0.70710678118654752440f));
}

__device__ __forceinline__ void atomicMaxFloat(float* addr, float val) {
    int* ia = reinterpret_cast<int*>(addr);
    int old = *ia;
    while (__int_as_float(old) < val) {
        int assumed = old;
        old = atomicCAS(ia, assumed, __float_as_int(val));
        if (old == assumed) break;
    }
}

// ---------------------------------------------------------------------------
// C[N x M] = A[N x K] @ W[K x M] + bias  (row major), K % 4 == 0, M % 64 == 0.
// Block = 128 threads (4 waves); block tile 64 rows x 64 cols.
// LDS layout (pair-interleaved): sB[(k>>1)*128 + 2*c + (k&1)] = W[k][col0+c]
//   -> lane's B fragment {W[k][c], W[k+1][c]} is one aligned ds_load_b64.
// A frag (16x4): lane l, reg i : m = l&15, k = (l>>4)*2 + i
// B frag (4x16): lane l, reg i : n = l&15, k = (l>>4)*2 + i
// C frag (16x16): lane l, reg j : m = j + 8*(l>>4), n = l&15
// Out-of-range A rows are clamped (never zeroed): a WMMA A-row only feeds the
// matching C row, and those C rows are never stored.
// ---------------------------------------------------------------------------
__global__ void wmma_gemm_bias(const float* __restrict__ A,
                               const float* __restrict__ W,
                               const float* __restrict__ bias,
                               float* __restrict__ C,
                               int N, int K, int M) {
    __shared__ float sB[256 * 64];            // 64 KB max (K <= 256)
    const int tid  = threadIdx.x;
    const int col0 = blockIdx.x * 64;

    // cooperative staging: pair p = k/2, col c; coalesced b32 reads,
    // one ds_store_b64 per (p, c)
    for (int idx = tid; idx < (K >> 1) * 64; idx += 128) {
        const int p = idx >> 6;
        const int c = idx & 63;
        const float w0 = W[(long)(2 * p) * M + col0 + c];
        const float w1 = W[(long)(2 * p + 1) * M + col0 + c];
        *(v2f*)(sB + p * 128 + 2 * c) = (v2f){w0, w1};
    }
    __syncthreads();

    const int lane = tid & 31;
    const int wave = tid >> 5;
    const int row0 = blockIdx.y * 64 + wave * 16;
    if (row0 >= N) return;

    const int mrow = lane & 15;
    const int kh   = (lane >> 4) * 2;         // 0 or 2

    int arow = row0 + mrow;
    if (arow >= N) arow = N - 1;              // clamp: keep reads in-bounds
    const float* Arow = A + (long)arow * K;

    // lane's LDS base: pair offset for kh plus 2*col-within-tile
    const float* sBl = sB + (lane >> 4) * 128 + 2 * mrow;

    v8f c0 = {}, c1 = {}, c2 = {}, c3 = {};
    v2f a_cur = *(const v2f*)(Arow + kh);
    for (int kk = 0; kk < K; kk += 4) {
        int kn = kk + 4;
        if (kn > K - 4) kn = K - 4;           // redundant last prefetch, in-bounds
        const v2f a_next = *(const v2f*)(Arow + kn + kh);

        const float* bp = sBl + kk * 64;      // (kk/2)*128
        const v2f b0 = *(const v2f*)(bp +  0);
        const v2f b1 = *(const v2f*)(bp + 32);
        const v2f b2 = *(const v2f*)(bp + 64);
        const v2f b3 = *(const v2f*)(bp + 96);

        c0 = __builtin_amdgcn_wmma_f32_16x16x4_f32(false, a_cur, false, b0, (short)0, c0, false, false);
        c1 = __builtin_amdgcn_wmma_f32_16x16x4_f32(false, a_cur, false, b1, (short)0, c1, false, false);
        c2 = __builtin_amdgcn_wmma_f32_16x16x4_f32(false, a_cur, false, b2, (short)0, c2, false, false);
        c3 = __builtin_amdgcn_wmma_f32_16x16x4_f32(false, a_cur, false, b3, (short)0, c3, false, false);
        a_cur = a_next;
    }

    const int rbase = row0 + ((lane >> 4) << 3);
    v8f acc[4] = {c0, c1, c2, c3};
#pragma unroll
    for (int t = 0; t < 4; ++t) {
        const int colw = col0 + t * 16 + mrow;
        const float bv = bias[colw];
#pragma unroll
        for (int j = 0; j < 8; ++j) {
            const int r = rbase + j;
            if (r < N) C[(long)r * M + colw] = acc[t][j] + bv;
        }
    }
}

// ---------------------------------------------------------------------------
__global__ void fill_kernel(float* __restrict__ p, float v, int n) {
    int i = blockIdx.x * blockDim.x + threadIdx.x;
    if (i < n) p[i] = v;
}

__global__ void gelu_kernel(float* __restrict__ p, long n) {
    const long i = (long)blockIdx.x * blockDim.x + threadIdx.x;
    if (i < n) p[i] = gelu_exact(p[i]);
}

// Pass 1: per (edge, head) logit a = (q[dst] . (k[src] @ Krel)) * p * scale,
// running segment max via CAS-max.
__global__ void edge_logits_kernel(const int* __restrict__ src,
                                   const int* __restrict__ dst, int E,
                                   const float* __restrict__ kqv,
                                   int srcBase, int dstBase,
                                   const float* __restrict__ krel,   // [4][16][16]
                                   const float* __restrict__ prel,   // [4]
                                   float* __restrict__ elog,         // [E][4]
                                   float* __restrict__ mbuf) {       // [Ntot][4]
    __shared__ float sk[1024];
    for (int i = threadIdx.x; i < 1024; i += blockDim.x) sk[i] = krel[i];
    __syncthreads();

    const int idx = blockIdx.x * blockDim.x + threadIdx.x;
    if (idx >= E * 4) return;
    const int e = idx >> 2, h = idx & 3;
    const int s = src[e], d = dst[e];

    const float* kp = kqv + (long)(srcBase + s) * 192 + h * 16;        // k
    const float* qp = kqv + (long)(dstBase + d) * 192 + 64 + h * 16;   // q
    const float* km = sk + h * 256;

    float acc = 0.0f;
#pragma unroll
    for (int o = 0; o < 16; ++o) {
        float kr = 0.0f;
#pragma unroll
        for (int dd = 0; dd < 16; ++dd) kr += kp[dd] * km[dd * 16 + o];
        acc += qp[o] * kr;
    }
    const float a = acc * prel[h] * 0.25f;   // scale = 1/sqrt(16)
    elog[(long)e * 4 + h] = a;
    atomicMaxFloat(&mbuf[(long)(dstBase + d) * 4 + h], a);
}

// Pass 2: e = exp(a - m[dst]); z[dst] += e (in-place overwrite of logits).
__global__ void edge_exp_kernel(const int* __restrict__ dst, int E, int dstBase,
                                float* __restrict__ elog,
                                const float* __restrict__ mbuf,
                                float* __restrict__ zbuf) {
    const int idx = blockIdx.x * blockDim.x + threadIdx.x;
    if (idx >= E * 4) return;
    const int e = idx >> 2, h = idx & 3;
    const int d = dst[e];
    const float ex = __expf(elog[idx] - mbuf[(long)(dstBase + d) * 4 + h]);
    elog[idx] = ex;
    atomicAdd(&zbuf[(long)(dstBase + d) * 4 + h], ex);
}

// Pass 3: alpha = e/(z+eps); agg[dst] += alpha * (v[src] @ Vrel).
__global__ void edge_agg_kernel(const int* __restrict__ src,
                                const int* __restrict__ dst, int E,
                                const float* __restrict__ kqv,
                                int srcBase, int dstBase,
                                const float* __restrict__ vrel,   // [4][16][16]
                                const float* __restrict__ elog,
                                const float* __restrict__ zbuf,
                                float* __restrict__ agg) {        // [Ntot][64]
    __shared__ float sv[1024];
    for (int i = threadIdx.x; i < 1024; i += blockDim.x) sv[i] = vrel[i];
    __syncthreads();

    const int idx = blockIdx.x * blockDim.x + threadIdx.x;
    if (idx >= E * 4) return;
    const int e = idx >> 2, h = idx & 3;
    const int s = src[e], d = dst[e];

    const float alpha = elog[idx] / (zbuf[(long)(dstBase + d) * 4 + h] + 1e-16f);
    const float* vp = kqv + (long)(srcBase + s) * 192 + 128 + h * 16;
    float* ap = agg + (long)(dstBase + d) * 64 + h * 16;
    const float* vm = sv + h * 256;
#pragma unroll
    for (int o = 0; o < 16; ++o) {
        float vr = 0.0f;
#pragma unroll
        for (int dd = 0; dd < 16; ++dd) vr += vp[dd] * vm[dd * 16 + o];
        atomicAdd(&ap[o], alpha * vr);
    }
}

// relu + zero-pad 64 -> 256 into h for the next layer.
__global__ void pad_relu_kernel(const float* __restrict__ o2,
                                float* __restrict__ h, long n256) {
    const long i = (long)blockIdx.x * blockDim.x + threadIdx.x;
    if (i >= n256) return;
    const long node = i >> 8;
    const int c = (int)(i & 255);
    h[i] = (c < 64) ? fmaxf(o2[node * 64 + c], 0.0f) : 0.0f;
}

// Final per-type projection to 2 channels (too narrow for WMMA).
__global__ void linout_kernel(const float* __restrict__ h,
                              const float* __restrict__ W,   // [256][2]
                              const float* __restrict__ b,   // [2]
                              float* __restrict__ out, int N) {
    const int n = blockIdx.x * blockDim.x + threadIdx.x;
    if (n >= N) return;
    const float* hp = h + (long)n * 256;
    float s0 = b[0], s1 = b[1];
    for (int k = 0; k < 256; ++k) {
        const float hv = hp[k];
        s0 += hv * W[k * 2];
        s1 += hv * W[k * 2 + 1];
    }
    out[(long)n * 2]     = s0;
    out[(long)n * 2 + 1] = s1;
}

// ---------------------------------------------------------------------------
// Host side: params pytree flat layout (JAX: dict keys sorted, tuples ordered)
// ---------------------------------------------------------------------------
extern "C" void kernel_launch(void* const* d_in, const int* in_sizes, int n_in,
                              void* d_out, int out_size, void* d_ws, size_t ws_size,
                              hipStream_t stream) {
    // NODE_TYPES order: person, device, session, utterance, event, entity, intent
    static const int   nN[7]        = {20000, 25000, 30000, 60000, 60000, 50000, 500};
    static const int   nBase[7]     = {0, 20000, 45000, 75000, 135000, 195000, 245000};
    static const int   sortedPos[7] = {4, 0, 5, 6, 2, 1, 3};   // -> alphabetical slot
    static const long  outOff[7]    = {0, 40000, 90000, 150000, 270000, 390000, 490000};
    static const long  lininCum[7]  = {0, 8448, 25088, 49920, 54272, 70912, 87552}; // by sorted slot

    // EDGE_ARG order
    static const int eSrcNt[10]  = {0, 1, 2, 3, 2, 4, 4, 3, 3, 5};
    static const int eDstNt[10]  = {1, 0, 3, 2, 4, 5, 4, 6, 5, 5};
    static const int eSorted[10] = {4, 0, 5, 9, 6, 2, 3, 7, 8, 1}; // -> alphabetical slot

    const long LAYER_SIZE  = 395048;
    const long KREL_OFF    = 0;        // + sortedEdge*1024
    const long KQV_OFF     = 10240;    // + sortedNt*49344  (b:192, w:256x192)
    const long OUT_OFF     = 355648;   // + sortedNt*4160   (b:64,  w:64x64)
    const long PREL_OFF    = 384768;   // + sortedEdge*4
    const long VREL_OFF    = 384808;   // + sortedEdge*1024
    const long LININ_BASE  = 790096;   // per sorted nt: b:256, w:IN_CH x 256
    const long LINOUT_BASE = 906576;   // + sortedNt*514    (b:2,   w:256x2)

    const float* P = (const float*)d_in[0];
    const float* X[7];
    for (int t = 0; t < 7; ++t) X[t] = (const float*)d_in[1 + t];

    const int* eSrc[10]; const int* eDst[10]; int eCnt[10]; long eBase[10];
    long ecum = 0;
    for (int j = 0; j < 10; ++j) {
        const int E = in_sizes[8 + j] / 2;
        eCnt[j]  = E;
        eSrc[j]  = (const int*)d_in[8 + j];
        eDst[j]  = eSrc[j] + E;
        eBase[j] = ecum;
        ecum    += E;
    }

    const long NTOT = 245500;
    // workspace partition (floats)
    float* ws   = (float*)d_ws;
    float* hbuf = ws;                            // NTOT*256
    float* kqvb = hbuf + NTOT * 256;             // NTOT*192
    float* aggb = kqvb + NTOT * 192;             // NTOT*64
    float* o2b  = aggb + NTOT * 64;              // NTOT*64
    float* mbuf = o2b  + NTOT * 64;              // NTOT*4
    float* zbuf = mbuf + NTOT * 4;               // NTOT*4
    float* elog = zbuf + NTOT * 4;               // ecum*4

    const dim3 blk128(128), blk256(256);

    // ---- lin_in: h[nt] = x @ W + b (M=256) ----
    for (int t = 0; t < 7; ++t) {
        const long off = LININ_BASE + lininCum[sortedPos[t]];
        const int  K   = in_sizes[1 + t] / nN[t];
        dim3 grid(256 / 64, (nN[t] + 63) / 64);
        wmma_gemm_bias<<<grid, blk128, 0, stream>>>(
            X[t], P + off + 256, P + off,
            hbuf + (long)nBase[t] * 256, nN[t], K, 256);
    }

    // ---- HGT conv layers ----
    for (int L = 0; L < 2; ++L) {
        const long lb = (long)L * LAYER_SIZE;

        // fused KQV projection per node type (M=192)
        for (int t = 0; t < 7; ++t) {
            const long off = lb + KQV_OFF + (long)sortedPos[t] * 49344;
            dim3 grid(192 / 64, (nN[t] + 63) / 64);
            wmma_gemm_bias<<<grid, blk128, 0, stream>>>(
                hbuf + (long)nBase[t] * 256, P + off + 192, P + off,
                kqvb + (long)nBase[t] * 192, nN[t], 256, 192);
        }

        // init segment max / sum / aggregate
        fill_kernel<<<(int)((NTOT * 4 + 255) / 256), blk256, 0, stream>>>(mbuf, -FLT_MAX, (int)(NTOT * 4));
        fill_kernel<<<(int)((NTOT * 4 + 255) / 256), blk256, 0, stream>>>(zbuf, 0.0f, (int)(NTOT * 4));
        fill_kernel<<<(int)((NTOT * 64 + 255) / 256), blk256, 0, stream>>>(aggb, 0.0f, (int)(NTOT * 64));

        // pass 1: logits + segment max (across ALL edge types before pass 2)
        for (int j = 0; j < 10; ++j) {
            dim3 grid((eCnt[j] * 4 + 255) / 256);
            edge_logits_kernel<<<grid, blk256, 0, stream>>>(
                eSrc[j], eDst[j], eCnt[j], kqvb,
                nBase[eSrcNt[j]], nBase[eDstNt[j]],
                P + lb + KREL_OFF + (long)eSorted[j] * 1024,
                P + lb + PREL_OFF + (long)eSorted[j] * 4,
                elog + eBase[j] * 4, mbuf);
        }
        // pass 2: exp + segment sum
        for (int j = 0; j < 10; ++j) {
            dim3 grid((eCnt[j] * 4 + 255) / 256);
            edge_exp_kernel<<<grid, blk256, 0, stream>>>(
                eDst[j], eCnt[j], nBase[eDstNt[j]],
                elog + eBase[j] * 4, mbuf, zbuf);
        }
        // pass 3: alpha-weighted V_rel scatter-add
        for (int j = 0; j < 10; ++j) {
            dim3 grid((eCnt[j] * 4 + 255) / 256);
            edge_agg_kernel<<<grid, blk256, 0, stream>>>(
                eSrc[j], eDst[j], eCnt[j], kqvb,
                nBase[eSrcNt[j]], nBase[eDstNt[j]],
                P + lb + VREL_OFF + (long)eSorted[j] * 1024,
                elog + eBase[j] * 4, zbuf, aggb);
        }

        // exact GELU in-place on agg, then plain out projection (M=K=64)
        gelu_kernel<<<(int)((NTOT * 64 + 255) / 256), blk256, 0, stream>>>(aggb, NTOT * 64);
        for (int t = 0; t < 7; ++t) {
            const long off = lb + OUT_OFF + (long)sortedPos[t] * 4160;
            dim3 grid(1, (nN[t] + 63) / 64);
            wmma_gemm_bias<<<grid, blk128, 0, stream>>>(
                aggb + (long)nBase[t] * 64, P + off + 64, P + off,
                o2b + (long)nBase[t] * 64, nN[t], 64, 64);
        }

        // relu + zero-pad back to 256 wide
        pad_relu_kernel<<<(int)((NTOT * 256 + 255) / 256), blk256, 0, stream>>>(
            o2b, hbuf, NTOT * 256);
    }

    // ---- lin_out per node type (2 channels) ----
    float* out = (float*)d_out;
    for (int t = 0; t < 7; ++t) {
        const long off = LINOUT_BASE + (long)sortedPos[t] * 514;
        linout_kernel<<<(nN[t] + 255) / 256, blk256, 0, stream>>>(
            hbuf + (long)nBase[t] * 256, P + off + 2, P + off,
            out + outOff[t], nN[t]);
    }
}